// PCENTransform_73014444032787
// MI455X (gfx1250) — compile-verified
//
#include <hip/hip_runtime.h>
#include <hip/hip_bf16.h>

// PCEN: per-row EMA over time (T=4096) + fused power-law compression.
// x: [B=32, F=256, T=4096] f32. One wave32 per row. Input is streamed with the
// CDNA5 async data mover (GLOBAL_LOAD_ASYNC_TO_LDS_B128, ASYNCcnt) into a
// 4-deep rotating LDS buffer, 2 chunks in flight. The EMA (constant-decay
// linear recurrence) is a geometric-weighted prefix sum via a DPP row_shr
// Hillis-Steele scan + v_readlane cross-row/carry broadcast (no LDS on the
// serial chain). Output is a write-once non-temporal b128 stream.

typedef float v4f __attribute__((ext_vector_type(4)));

#define T_DIM 4096
#define CHUNK 128                 // 32 lanes * 4 elems
#define NCHUNK (T_DIM / CHUNK)    // 32
#define WAVES_PER_BLOCK 8         // 256 threads / 32
#define NBUF 4                    // LDS buffers per wave (rotation)

// DPP row_shr:d with bound_ctrl -> zero shifted in at 16-lane row boundaries.
#define DPPF_SHR(x, d)                                                        \
    __int_as_float(__builtin_amdgcn_update_dpp(                               \
        0, __float_as_int(x), 0x110 | (d), 0xF, 0xF, true))

#define READLANE_F(x, l)                                                      \
    __int_as_float(__builtin_amdgcn_readlane(__float_as_int(x), (l)))

// Async DMA: global -> LDS, 16B per lane, tracked by ASYNCcnt (GV mode).
__device__ __forceinline__ void async_ld_b128(unsigned lds_byte_off,
                                              const float* gaddr) {
    asm volatile("global_load_async_to_lds_b128 %0, %1, off"
                 :: "v"(lds_byte_off), "v"(gaddr)
                 : "memory");
}
#define WAIT_ASYNC_1() asm volatile("s_wait_asynccnt 0x1" ::: "memory")
#define WAIT_ASYNC_0() asm volatile("s_wait_asynccnt 0x0" ::: "memory")

__device__ __forceinline__ float pcen_elem(float x, float m) {
    // (x * (m+eps)^-alpha + delta)^0.5 - delta^0.5
    float l = __builtin_amdgcn_logf(m + 1e-6f);      // v_log_f32 (log2)
    float p = __builtin_amdgcn_exp2f(-0.98f * l);    // v_exp_f32 (exp2)
    float v = __builtin_fmaf(x, p, 2.0f);
    return __builtin_sqrtf(v) - 1.4142135623730951f;
}

__global__ __launch_bounds__(256)
void pcen_kernel(const float* __restrict__ x, float* __restrict__ out, int rows) {
    __shared__ __align__(16) float sbuf[WAVES_PER_BLOCK][NBUF][CHUNK]; // 16 KB

    const int lane = threadIdx.x & 31;
    const int wid  = threadIdx.x >> 5;
    const int row  = blockIdx.x * WAVES_PER_BLOCK + wid;
    if (row >= rows) return;   // wave-uniform; EXEC stays all-ones inside

    const float s  = 0.025f;
    const float r  = 0.975f;            // 1 - s
    const float r2 = r * r;
    const float r3 = r2 * r;
    const float r4 = r2 * r2;           // q    = r^4 (per-lane decay)
    const float w2 = r4 * r4;           // q^2  = r^8
    const float w4 = w2 * w2;           // q^4  = r^16
    const float w8 = w4 * w4;           // q^8  = r^32

    // Per-lane constants.
    const float lq = __builtin_amdgcn_logf(r4);                        // log2(q)
    const float pow4lane = __builtin_amdgcn_exp2f((float)lane * lq);   // q^lane
    const float rowfix   = (lane >= 16)
        ? __builtin_amdgcn_exp2f((float)(lane - 15) * lq) : 0.0f;      // q^(lane-15)
    const float mask16   = (lane == 16) ? 1.0f : 0.0f;

    const float* xp = x   + (size_t)row * T_DIM + (size_t)lane * 4;
    float*       op = out + (size_t)row * T_DIM + (size_t)lane * 4;

    // Per-lane LDS byte offsets of the 4 buffers (flat->LDS: low 32 bits).
    unsigned lds_off[NBUF];
#pragma unroll
    for (int b = 0; b < NBUF; ++b)
        lds_off[b] = (unsigned)(uintptr_t)&sbuf[wid][b][lane * 4];

    float carry = 0.0f;  // m at last sample of previous chunk (uniform scalar)

    // Process one 128-sample chunk from LDS buffer b; advance carry; store.
    auto process = [&](int b, float* o) {
        v4f xv = *(const v4f*)&sbuf[wid][b][lane * 4];   // ds_load_b128

        // lane-local weighted inclusive prefix: y_k = x_k + r*y_{k-1}
        float y0 = xv.x;
        float y1 = __builtin_fmaf(r, y0, xv.y);
        float y2 = __builtin_fmaf(r, y1, xv.z);
        float y3 = __builtin_fmaf(r, y2, xv.w);

        // DPP weighted scan over lane aggregates within 16-lane rows
        float A = y3;
        A = __builtin_fmaf(r4, DPPF_SHR(A, 1), A);
        A = __builtin_fmaf(w2, DPPF_SHR(A, 2), A);
        A = __builtin_fmaf(w4, DPPF_SHR(A, 4), A);
        A = __builtin_fmaf(w8, DPPF_SHR(A, 8), A);

        // cross-row fix: row-0 aggregate into lanes 16..31 (scalar readlane)
        float upper = READLANE_F(A, 15);
        A = __builtin_fmaf(rowfix, upper, A);

        // exclusive prefix: shift by 1; lane16 gets row-0 aggregate
        float E = __builtin_fmaf(mask16, upper, DPPF_SHR(A, 1));

        // m_k = s*y_k + r^{k+1} * (s*E + carry*q^lane)
        float g  = __builtin_fmaf(s, E, carry * pow4lane);
        float m0 = __builtin_fmaf(r,  g, s * y0);
        float m1 = __builtin_fmaf(r2, g, s * y1);
        float m2 = __builtin_fmaf(r3, g, s * y2);
        float m3 = __builtin_fmaf(r4, g, s * y3);

        carry = READLANE_F(m3, 31);   // v_readlane -> SGPR scalar

        v4f ov;
        ov.x = pcen_elem(xv.x, m0);
        ov.y = pcen_elem(xv.y, m1);
        ov.z = pcen_elem(xv.z, m2);
        ov.w = pcen_elem(xv.w, m3);
        __builtin_nontemporal_store(ov, (v4f*)o);   // write-once NT stream
    };

    // Prologue: 2 chunks in flight on the async engine.
    async_ld_b128(lds_off[0], xp + 0 * CHUNK);
    async_ld_b128(lds_off[1], xp + 1 * CHUNK);

    // Main loop: 4x unrolled so buffer indices / wait immediates are constants.
    // Chunks 0..27 (7 iterations); at chunk c we issue chunk c+2.
    for (int c = 0; c < NCHUNK - NBUF; c += NBUF) {
#pragma unroll
        for (int u = 0; u < NBUF; ++u) {
            WAIT_ASYNC_1();                                  // chunk c+u landed
            async_ld_b128(lds_off[(u + 2) & (NBUF - 1)],
                          xp + (size_t)(c + u + 2) * CHUNK); // issue c+u+2
            process(u, op);
            op += CHUNK;
        }
    }
    // Tail: chunks 28..31 (issues for 30,31 happen here; then drain).
    WAIT_ASYNC_1(); async_ld_b128(lds_off[2], xp + (size_t)(NCHUNK - 2) * CHUNK);
    process(0, op); op += CHUNK;
    WAIT_ASYNC_1(); async_ld_b128(lds_off[3], xp + (size_t)(NCHUNK - 1) * CHUNK);
    process(1, op); op += CHUNK;
    WAIT_ASYNC_1(); process(2, op); op += CHUNK;
    WAIT_ASYNC_0(); process(3, op);
}

extern "C" void kernel_launch(void* const* d_in, const int* in_sizes, int n_in,
                              void* d_out, int out_size, void* d_ws, size_t ws_size,
                              hipStream_t stream) {
    (void)n_in; (void)out_size; (void)d_ws; (void)ws_size;
    const float* x = (const float*)d_in[0];
    float* out = (float*)d_out;

    const int rows = in_sizes[0] / T_DIM;               // 32*256 = 8192
    const int blocks = (rows + WAVES_PER_BLOCK - 1) / WAVES_PER_BLOCK;  // 1024

    pcen_kernel<<<blocks, 256, 0, stream>>>(x, out, rows);
}